// SpikeFP8MulToFP32_43860206027269
// MI455X (gfx1250) — compile-verified
//
#include <hip/hip_runtime.h>
#include <cstdint>
#include <cstddef>

typedef unsigned int u32;
typedef unsigned long long u64;
typedef u32 __attribute__((ext_vector_type(4))) u32x4;

#define BLOCK 256
#define RPT 8  // rows per thread

// Flat (generic) pointers to LDS carry the wave-relative LDS byte offset in
// their low 32 bits on gfx1250 (ISA 10.2).
__device__ __forceinline__ u32 lds_off(const void* p) {
  return (u32)(uintptr_t)p;
}

// Issue async copies of one 32B A-row and one 32B B-row into this thread's
// LDS slot (A at +0..31, B at +32..63). 4 x global_load_async_to_lds_b128,
// tracked with ASYNCcnt. The instruction offset is added to BOTH the LDS and
// the global address, so offset:16 fetches the second half of each row.
__device__ __forceinline__ void stage_rows(u32 ldsBase, const float* Arow,
                                           const float* Brow) {
  u64 ga = (u64)(uintptr_t)Arow;
  u64 gb = (u64)(uintptr_t)Brow;
  u32 ldsA = ldsBase;
  u32 ldsB = ldsBase + 32u;
  asm volatile(
      "global_load_async_to_lds_b128 %0, %1, off\n\t"
      "global_load_async_to_lds_b128 %0, %1, off offset:16"
      :: "v"(ldsA), "v"(ga) : "memory");
  asm volatile(
      "global_load_async_to_lds_b128 %0, %1, off\n\t"
      "global_load_async_to_lds_b128 %0, %1, off offset:16"
      :: "v"(ldsB), "v"(gb) : "memory");
}

// Input floats are exactly 0.0f (0x00000000) or 1.0f (0x3F800000):
// bit 29 of the fp32 pattern is the value.
__device__ __forceinline__ u32 bit_of(u32 fbits) { return (fbits >> 29) & 1u; }

__device__ __forceinline__ u32 pack_e4m3(u32x4 lo, u32x4 hi) {
  u32 r;
  r  = bit_of(lo.x) << 7;   // S
  r |= bit_of(lo.y) << 6;   // E3
  r |= bit_of(lo.z) << 5;   // E2
  r |= bit_of(lo.w) << 4;   // E1
  r |= bit_of(hi.x) << 3;   // E0
  r |= bit_of(hi.y) << 2;   // M2
  r |= bit_of(hi.z) << 1;   // M1
  r |= bit_of(hi.w);        // M0
  return r;
}

// Decode per the reference circuit (no NaN/Inf path; e=15,m=7 -> 480):
//   e>0: (1+m/8)*2^(e-7)  -> fp32 pattern exp=e+120, mant=m<<20
//   e==0: (m/8)*2^-6 = m*2^-9 (exact fp32); sign OR'd in (gives -0.0 like ref)
__device__ __forceinline__ float decode_e4m3(u32 c) {
  u32 s = (c >> 7) & 1u;
  u32 e = (c >> 3) & 0xFu;
  u32 m = c & 0x7u;
  u32 norm = ((e + 120u) << 23) | (m << 20);
  u32 sub  = __float_as_uint((float)m * 0x1p-9f);
  u32 mag  = (e != 0u) ? norm : sub;
  return __uint_as_float(mag | (s << 31));
}

__global__ __launch_bounds__(BLOCK) void fp8mul_bits_kernel(
    const float* __restrict__ A, const float* __restrict__ B,
    float* __restrict__ out, long long nrows) {
  // [2 buffers][BLOCK threads][4 x uint4 = A.lo, A.hi, B.lo, B.hi] = 32 KiB
  __shared__ u32x4 smem[2 * BLOCK * 4];

  const int tid = threadIdx.x;
  const long long NT = (long long)gridDim.x * BLOCK;
  const long long row0 = (long long)blockIdx.x * BLOCK + tid;

  const u32 base0 = lds_off(&smem[tid * 4]);
  const u32 bufStride = BLOCK * 4 * sizeof(u32x4);  // 16 KiB between buffers

  // Prologue: stage tile 0 into buffer 0 (clamp OOB rows to row 0: safe read).
  {
    long long r = (row0 < nrows) ? row0 : 0;
    stage_rows(base0, A + r * 8, B + r * 8);
  }

#pragma unroll
  for (int k = 0; k < RPT; ++k) {
    const long long row = row0 + (long long)k * NT;
    const int cur = k & 1;

    if (k + 1 < RPT) {
      // Kick off the next tile into the other buffer, then wait for the
      // 4 oldest async ops (this tile). Async loads complete in order.
      long long nr = row + NT;
      long long rs = (nr < nrows) ? nr : 0;
      stage_rows(base0 + (u32)(cur ^ 1) * bufStride, A + rs * 8, B + rs * 8);
      asm volatile("s_wait_asynccnt 0x4" ::: "memory");
    } else {
      asm volatile("s_wait_asynccnt 0x0" ::: "memory");
    }

    if (row < nrows) {
      const int idx = (cur * BLOCK + tid) * 4;
      u32x4 aLo = smem[idx + 0];
      u32x4 aHi = smem[idx + 1];
      u32x4 bLo = smem[idx + 2];
      u32x4 bHi = smem[idx + 3];

      float av = decode_e4m3(pack_e4m3(aLo, aHi));
      float bv = decode_e4m3(pack_e4m3(bLo, bHi));
      const u32 p = __float_as_uint(av * bv);  // exact fp32, matches reference

      u32x4* orow = (u32x4*)(out + row * 32);
#pragma unroll
      for (int q = 0; q < 8; ++q) {
        u32x4 o;
        o.x = ((p >> (31 - (4 * q + 0))) & 1u) * 0x3F800000u;
        o.y = ((p >> (31 - (4 * q + 1))) & 1u) * 0x3F800000u;
        o.z = ((p >> (31 - (4 * q + 2))) & 1u) * 0x3F800000u;
        o.w = ((p >> (31 - (4 * q + 3))) & 1u) * 0x3F800000u;
        __builtin_nontemporal_store(o, orow + q);  // streaming 512 MiB output
      }
    }
  }
}

extern "C" void kernel_launch(void* const* d_in, const int* in_sizes, int n_in,
                              void* d_out, int out_size, void* d_ws,
                              size_t ws_size, hipStream_t stream) {
  const float* A = (const float*)d_in[0];
  const float* B = (const float*)d_in[1];
  float* out = (float*)d_out;

  const long long nrows = (long long)in_sizes[0] / 8;  // 4,194,304
  long long threads_needed = (nrows + RPT - 1) / RPT;
  int blocks = (int)((threads_needed + BLOCK - 1) / BLOCK);
  if (blocks < 1) blocks = 1;

  fp8mul_bits_kernel<<<blocks, BLOCK, 0, stream>>>(A, B, out, nrows);
}